// GpuWaveletDenoiser_7816840479021
// MI455X (gfx1250) — compile-verified
//
// GpuWaveletDenoiser for MI455X (gfx1250).
// Bandwidth-bound: 512 MB traffic -> ~22 us floor @ 23.3 TB/s. Median/sort is
// the only real compute; done cooperatively in LDS. Data movement uses the
// CDNA5 async global<->LDS DMA path (ASYNCcnt) with non-temporal hints
// (streaming data, zero reuse -> don't churn the 192 MB L2).
#include <hip/hip_runtime.h>
#include <cstdint>
#include <math.h>

#ifndef __has_builtin
#define __has_builtin(x) 0
#endif

constexpr int kT = 512;        // signal length
constexpr int kF = 32;         // features per sample (row width in floats)
constexpr int kFH = 16;        // features (signals) handled per workgroup
constexpr int kThreads = 256;  // 8 wave32 per workgroup
constexpr float kSqrt2 = 0.70710678118654752440f;
constexpr int kCpolNT = 1;     // CPol: TH[2:0] = 1 (non-temporal)

// dynamic LDS layout (float offsets), total 23056 floats = 92224 B (~90 KB)
constexpr int oRaw = 0;        // 512 x 16 tile (in-place blended for output)
constexpr int oCA1 = 8192;     // 256 x 16  (reused as sort scratch)
constexpr int oCD1 = 12288;    // 256 x 16
constexpr int oCA2 = 16384;    // 128 x 16
constexpr int oCD2 = 18432;    // 128 x 16
constexpr int oCA3 = 20480;    //  64 x 16
constexpr int oCD3 = 21504;    //  64 x 16
constexpr int oRedS = 22528;   //  16 x 16 partial sums
constexpr int oRedQ = 22784;   //  16 x 16 partial sum of squares
constexpr int oThr = 23040;    //  16 thresholds
constexpr int kSmemFloats = 23056;
constexpr size_t kSmemBytes = kSmemFloats * sizeof(float);

// Async DMA builtins take int4 pointers: (v4i AS1* gaddr, v4i AS3* lds, Ii, Ii)
typedef int v4i __attribute__((vector_size(16)));
typedef __attribute__((address_space(1))) v4i* as1_v4i;
typedef __attribute__((address_space(3))) v4i* as3_v4i;

#if __has_builtin(__builtin_amdgcn_global_load_async_to_lds_b128)
#define USE_ASYNC_LOAD 1
#else
#define USE_ASYNC_LOAD 0
#endif
#if __has_builtin(__builtin_amdgcn_global_store_async_from_lds_b128)
#define USE_ASYNC_STORE 1
#else
#define USE_ASYNC_STORE 0
#endif

__device__ __forceinline__ void wait_asynccnt0() {
#if __has_builtin(__builtin_amdgcn_s_wait_asynccnt)
  __builtin_amdgcn_s_wait_asynccnt(0);
#else
  asm volatile("s_wait_asynccnt 0" ::: "memory");
#endif
}

// Generic->AS1: same 64-bit representation. Generic->AS3: low 32 bits hold the
// LDS offset (aperture layout), so integer truncation is the addrspacecast.
__device__ __forceinline__ as1_v4i to_global_v4(const void* p) {
  return (as1_v4i)(uintptr_t)p;  // also drops const (builtin wants non-const)
}
__device__ __forceinline__ as3_v4i to_lds_v4(void* p) {
  return (as3_v4i)(uint32_t)(uintptr_t)p;
}

// Sort 16 independent columns (stride kFH) of 256 elements, ascending.
// Bank-conflict-free: each half-wave touches 16 distinct banks per access.
__device__ __forceinline__ void bitonic_sort_cols(float* __restrict__ arr,
                                                  int f, int sub) {
  for (int k = 2; k <= 256; k <<= 1) {
    for (int j = k >> 1; j > 0; j >>= 1) {
      for (int i = sub; i < 256; i += 16) {
        const int ixj = i ^ j;
        if (ixj > i) {
          const int ia = i * kFH + f;
          const int ib = ixj * kFH + f;
          const float a = arr[ia];
          const float b = arr[ib];
          const bool asc = ((i & k) == 0);
          if (asc ? (a > b) : (a < b)) {
            arr[ia] = b;
            arr[ib] = a;
          }
        }
      }
      __syncthreads();
    }
  }
}

__global__ __launch_bounds__(kThreads) void haar_denoise_kernel(
    const float* __restrict__ x, float* __restrict__ out) {
  extern __shared__ float smem[];
  const int tid = threadIdx.x;
  const int f = tid & (kFH - 1);   // signal (column) id within tile
  const int sub = tid >> 4;        // 0..15: worker id within a signal
  const int n = blockIdx.x >> 1;
  const int half = blockIdx.x & 1;

  const float* gin = x + (size_t)n * (kT * kF) + half * kFH;
  float* gout = out + (size_t)n * (kT * kF) + half * kFH;

  // ---- Phase A: stage tile (512 rows x 16 floats, global row stride 32) ----
#if USE_ASYNC_LOAD
#pragma unroll
  for (int i = 0; i < 8; ++i) {
    const int c = i * kThreads + tid;  // 16-byte chunk id, 0..2047
    const int t = c >> 2;
    const int j4 = (c & 3) << 2;
    __builtin_amdgcn_global_load_async_to_lds_b128(
        to_global_v4(gin + t * kF + j4), to_lds_v4(smem + oRaw + (c << 2)), 0,
        kCpolNT);
  }
  wait_asynccnt0();
#else
  for (int i = 0; i < 8; ++i) {
    const int c = i * kThreads + tid;
    const int t = c >> 2;
    const int j4 = (c & 3) << 2;
    const float4 v = *reinterpret_cast<const float4*>(gin + t * kF + j4);
    *reinterpret_cast<float4*>(smem + oRaw + (c << 2)) = v;
  }
#endif
  __syncthreads();

  // ---- Phase B: forward Haar DWT, 3 levels ----
  for (int idx = tid; idx < 4096; idx += kThreads) {  // level 1: 256 k x 16 f
    const int k = idx >> 4;
    const float a = smem[oRaw + (2 * k) * kFH + f];
    const float b = smem[oRaw + (2 * k + 1) * kFH + f];
    smem[oCA1 + idx] = (a + b) * kSqrt2;
    smem[oCD1 + idx] = (a - b) * kSqrt2;
  }
  __syncthreads();
  for (int idx = tid; idx < 2048; idx += kThreads) {  // level 2
    const int k = idx >> 4;
    const float a = smem[oCA1 + (2 * k) * kFH + f];
    const float b = smem[oCA1 + (2 * k + 1) * kFH + f];
    smem[oCA2 + idx] = (a + b) * kSqrt2;
    smem[oCD2 + idx] = (a - b) * kSqrt2;
  }
  __syncthreads();
  for (int idx = tid; idx < 1024; idx += kThreads) {  // level 3
    const int k = idx >> 4;
    const float a = smem[oCA2 + (2 * k) * kFH + f];
    const float b = smem[oCA2 + (2 * k + 1) * kFH + f];
    smem[oCA3 + idx] = (a + b) * kSqrt2;
    smem[oCD3 + idx] = (a - b) * kSqrt2;
  }
  __syncthreads();

  // ---- Phase C: moments of cD1 + copy into sort scratch (cA1 is free now) --
  {
    float s0 = 0.f, s1 = 0.f;
    for (int idx = tid; idx < 4096; idx += kThreads) {
      const float v = smem[oCD1 + idx];
      smem[oCA1 + idx] = v;
      s0 += v;
      s1 += v * v;
    }
    smem[oRedS + tid] = s0;  // tid == sub*16 + f
    smem[oRedQ + tid] = s1;
  }
  __syncthreads();

  // ---- Phase D: median, MAD, threshold ----
  bitonic_sort_cols(smem + oCA1, f, sub);
  const float med = smem[oCA1 + 127 * kFH + f];  // lower median, k=(256-1)/2
  __syncthreads();
  for (int idx = tid; idx < 4096; idx += kThreads) {
    smem[oCA1 + idx] = fabsf(smem[oCD1 + idx] - med);
  }
  __syncthreads();
  bitonic_sort_cols(smem + oCA1, f, sub);
  if (tid < kFH) {
    float s = 0.f, q = 0.f;
#pragma unroll
    for (int w = 0; w < 16; ++w) {
      s += smem[oRedS + w * kFH + tid];
      q += smem[oRedQ + w * kFH + tid];
    }
    const float inv = 1.0f / 256.0f;
    const float mean = s * inv;
    const float var = fmaxf(q * inv - mean * mean, 0.0f);
    const float mad = smem[oCA1 + 127 * kFH + tid];
    const float sn = mad * (1.0f / 0.6745f);
    const float sn2 = sn * sn;
    const float ss = sqrtf(fmaxf(var - sn2, 1e-10f));
    smem[oThr + tid] = fmaxf(sn2 / ss * 0.3f, 0.0f);
  }
  __syncthreads();
  {
    const float tf = smem[oThr + f];
    for (int idx = tid; idx < 4096; idx += kThreads) {
      const float v = smem[oCD1 + idx];
      const float m = fmaxf(fabsf(v) - tf, 0.0f);
      smem[oCD1 + idx] = copysignf(m, v);  // sign(v)*max(|v|-thr,0)
    }
  }
  __syncthreads();

  // ---- Phase E: inverse DWT, blending into raw in place ----
  for (int idx = tid; idx < 1024; idx += kThreads) {  // (cA3,cD3) -> cA2
    const int k = idx >> 4;
    const float a = smem[oCA3 + idx];
    const float d = smem[oCD3 + idx];
    smem[oCA2 + (2 * k) * kFH + f] = (a + d) * kSqrt2;
    smem[oCA2 + (2 * k + 1) * kFH + f] = (a - d) * kSqrt2;
  }
  __syncthreads();
  for (int idx = tid; idx < 2048; idx += kThreads) {  // (cA2,cD2) -> cA1
    const int k = idx >> 4;
    const float a = smem[oCA2 + idx];
    const float d = smem[oCD2 + idx];
    smem[oCA1 + (2 * k) * kFH + f] = (a + d) * kSqrt2;
    smem[oCA1 + (2 * k + 1) * kFH + f] = (a - d) * kSqrt2;
  }
  __syncthreads();
  for (int idx = tid; idx < 4096; idx += kThreads) {  // final level + blend
    const int k = idx >> 4;
    const float a = smem[oCA1 + idx];
    const float d = smem[oCD1 + idx];
    const float ev = (a + d) * kSqrt2;
    const float od = (a - d) * kSqrt2;
    const int i0 = (2 * k) * kFH + f;
    const int i1 = i0 + kFH;
    smem[oRaw + i0] = 0.75f * smem[oRaw + i0] + 0.25f * ev;
    smem[oRaw + i1] = 0.75f * smem[oRaw + i1] + 0.25f * od;
  }
  __syncthreads();

  // ---- Phase F: stream tile out (LDS -> global DMA) ----
#if USE_ASYNC_STORE
#pragma unroll
  for (int i = 0; i < 8; ++i) {
    const int c = i * kThreads + tid;
    const int t = c >> 2;
    const int j4 = (c & 3) << 2;
    __builtin_amdgcn_global_store_async_from_lds_b128(
        to_global_v4(gout + t * kF + j4), to_lds_v4(smem + oRaw + (c << 2)), 0,
        kCpolNT);
  }
  wait_asynccnt0();
#else
  for (int i = 0; i < 8; ++i) {
    const int c = i * kThreads + tid;
    const int t = c >> 2;
    const int j4 = (c & 3) << 2;
    *reinterpret_cast<float4*>(gout + t * kF + j4) =
        *reinterpret_cast<const float4*>(smem + oRaw + (c << 2));
  }
#endif
}

extern "C" void kernel_launch(void* const* d_in, const int* in_sizes, int n_in,
                              void* d_out, int out_size, void* d_ws,
                              size_t ws_size, hipStream_t stream) {
  (void)n_in;
  (void)out_size;
  (void)d_ws;
  (void)ws_size;
  const float* x = reinterpret_cast<const float*>(d_in[0]);
  float* y = reinterpret_cast<float*>(d_out);
  const int N = in_sizes[0] / (kT * kF);  // 4096
  // ~90 KB dynamic LDS per workgroup (3 WGs / WGP of the 320 KB pool).
  (void)hipFuncSetAttribute(reinterpret_cast<const void*>(haar_denoise_kernel),
                            hipFuncAttributeMaxDynamicSharedMemorySize,
                            (int)kSmemBytes);
  const int nBlocks = N * (kF / kFH);  // 8192
  haar_denoise_kernel<<<dim3(nBlocks), dim3(kThreads), kSmemBytes, stream>>>(x,
                                                                             y);
}